// PointNet2SetAbstraction_81733227643069
// MI455X (gfx1250) — compile-verified
//
#include <hip/hip_runtime.h>
#include <hip/hip_bf16.h>

// ---------------------------------------------------------------------------
// PointNet++ SetAbstraction for MI455X (gfx1250, wave32, WMMA)
//   d_in: xyz[8,8192,3], features[8,64,8192],
//         W0[64,67],b0,g0,beta0, W1[64,64],b1,g1,beta1, W2[128,64],b2,g2,beta2
//   d_out: new_xyz[8,2048,3] ++ new_features[8,128,2048]  (f32)
// ---------------------------------------------------------------------------

#define BB     8
#define NN     8192
#define PP     2048
#define KK     32
#define CIN    64
#define R2     0.01f
#define INVSTD 0.9999950000374998f   // 1/sqrt(1+1e-5)

typedef float v2f __attribute__((ext_vector_type(2)));
typedef float v8f __attribute__((ext_vector_type(8)));

// ---------------------------------------------------------------------------
// Kernel 1: farthest point sampling. One block per batch, 1024 threads,
// 8 points per thread held in registers; two-level (shfl + LDS) argmax.
// ---------------------------------------------------------------------------
__global__ __launch_bounds__(1024) void fps_kernel(const float* __restrict__ xyz,
                                                   float* __restrict__ newxyz) {
  const int b = blockIdx.x;
  const int t = threadIdx.x;
  const int wid  = t >> 5;
  const int lane = t & 31;
  const int PT = NN / 1024;  // 8

  __shared__ float sv[32];
  __shared__ int   si[32];
  __shared__ float bc[3];

  const float* base = xyz + (size_t)b * NN * 3;
  const float c0x = base[0], c0y = base[1], c0z = base[2];

  float px[PT], py[PT], pz[PT], md[PT];
#pragma unroll
  for (int i = 0; i < PT; ++i) {
    const int n = t + i * 1024;
    px[i] = base[n * 3 + 0];
    py[i] = base[n * 3 + 1];
    pz[i] = base[n * 3 + 2];
    const float dx = px[i] - c0x, dy = py[i] - c0y, dz = pz[i] - c0z;
    md[i] = dx * dx + dy * dy + dz * dz;
  }
  if (t == 0) {
    newxyz[(size_t)b * PP * 3 + 0] = c0x;
    newxyz[(size_t)b * PP * 3 + 1] = c0y;
    newxyz[(size_t)b * PP * 3 + 2] = c0z;
  }

  for (int j = 1; j < PP; ++j) {
    // thread-local argmax (first max wins -> smallest index)
    float bv = -1.0f;
    int   bi = 0x7fffffff;
#pragma unroll
    for (int i = 0; i < PT; ++i) {
      if (md[i] > bv) { bv = md[i]; bi = t + i * 1024; }
    }
    // wave32 reduce
#pragma unroll
    for (int off = 16; off > 0; off >>= 1) {
      const float ov = __shfl_xor(bv, off, 32);
      const int   oi = __shfl_xor(bi, off, 32);
      if (ov > bv || (ov == bv && oi < bi)) { bv = ov; bi = oi; }
    }
    if (lane == 0) { sv[wid] = bv; si[wid] = bi; }
    __syncthreads();
    if (wid == 0) {
      bv = sv[lane];
      bi = si[lane];
#pragma unroll
      for (int off = 16; off > 0; off >>= 1) {
        const float ov = __shfl_xor(bv, off, 32);
        const int   oi = __shfl_xor(bi, off, 32);
        if (ov > bv || (ov == bv && oi < bi)) { bv = ov; bi = oi; }
      }
      if (lane == 0) {
        const float gx = base[bi * 3 + 0];
        const float gy = base[bi * 3 + 1];
        const float gz = base[bi * 3 + 2];
        bc[0] = gx; bc[1] = gy; bc[2] = gz;
        newxyz[((size_t)b * PP + j) * 3 + 0] = gx;
        newxyz[((size_t)b * PP + j) * 3 + 1] = gy;
        newxyz[((size_t)b * PP + j) * 3 + 2] = gz;
      }
    }
    __syncthreads();
    const float cx = bc[0], cy = bc[1], cz = bc[2];
#pragma unroll
    for (int i = 0; i < PT; ++i) {
      const float dx = px[i] - cx, dy = py[i] - cy, dz = pz[i] - cz;
      const float d2 = dx * dx + dy * dy + dz * dz;
      md[i] = fminf(md[i], d2);
    }
  }
}

// ---------------------------------------------------------------------------
// One 16xCOUTx... MLP layer on a 32-row tile using V_WMMA_F32_16X16X4_F32.
// Xs: [32 x 68] row-major in LDS (stride 68), Hs: output LDS tile (stride 68).
// ---------------------------------------------------------------------------
__device__ __forceinline__ void mlp_layer(const float* __restrict__ Xs,
                                          float* __restrict__ Hs,
                                          const float* __restrict__ W,
                                          const float* __restrict__ bconv,
                                          const float* __restrict__ gam,
                                          const float* __restrict__ beta,
                                          int cin, int cout, int lane) {
  const int chunks = (cin + 3) >> 2;
  const int m0  = lane & 15;
  const int kk0 = (lane >> 4) << 1;   // lanes 0-15: K=0,1 ; lanes 16-31: K=2,3
  for (int nt = 0; nt < (cout >> 4); ++nt) {
    const int n = (nt << 4) + (lane & 15);
    v8f acc0 = {};
    v8f acc1 = {};
    for (int c = 0; c < chunks; ++c) {
      const int c4 = c << 2;
      v2f a0, a1, bf;
      a0.x = Xs[m0 * 68 + c4 + kk0];
      a0.y = Xs[m0 * 68 + c4 + kk0 + 1];
      a1.x = Xs[(m0 + 16) * 68 + c4 + kk0];
      a1.y = Xs[(m0 + 16) * 68 + c4 + kk0 + 1];
      const int ca = c4 + kk0;
      bf.x = (ca     < cin) ? W[n * cin + ca]     : 0.0f;
      bf.y = (ca + 1 < cin) ? W[n * cin + ca + 1] : 0.0f;
      acc0 = __builtin_amdgcn_wmma_f32_16x16x4_f32(false, a0, false, bf,
                                                   (short)0, acc0, false, false);
      acc1 = __builtin_amdgcn_wmma_f32_16x16x4_f32(false, a1, false, bf,
                                                   (short)0, acc1, false, false);
    }
    const float s = gam[n] * INVSTD;
    const float tt = bconv[n] * s + beta[n];
#pragma unroll
    for (int r = 0; r < 8; ++r) {
      const int mrow = r + ((lane >> 4) << 3);
      const float v0 = fmaxf(acc0[r] * s + tt, 0.0f);
      const float v1 = fmaxf(acc1[r] * s + tt, 0.0f);
      Hs[mrow * 68 + (nt << 4) + (lane & 15)]        = v0;
      Hs[(mrow + 16) * 68 + (nt << 4) + (lane & 15)] = v1;
    }
  }
}

// ---------------------------------------------------------------------------
// Kernel 2 (fused): ball query + group + 3-layer MLP + max-pool.
// One wave32 per centroid; 4 waves per block; wave-private LDS tiles.
// ---------------------------------------------------------------------------
#define WAVES 4
__global__ __launch_bounds__(WAVES * 32) void sa_fused(
    const float* __restrict__ xyz, const float* __restrict__ feat,
    const float* __restrict__ W0, const float* __restrict__ b0,
    const float* __restrict__ g0, const float* __restrict__ bt0,
    const float* __restrict__ W1, const float* __restrict__ b1,
    const float* __restrict__ g1, const float* __restrict__ bt1,
    const float* __restrict__ W2, const float* __restrict__ b2,
    const float* __restrict__ g2, const float* __restrict__ bt2,
    const float* __restrict__ newxyz, float* __restrict__ outFeat) {
  __shared__ float buf[2][WAVES][32][68];
  __shared__ int   nbr[WAVES][32];

  const int w    = threadIdx.x >> 5;
  const int lane = threadIdx.x & 31;
  const int g    = blockIdx.x * WAVES + w;   // global centroid id
  const int b    = g >> 11;                  // / 2048
  const int p    = g & (PP - 1);

  const float cx = newxyz[(size_t)g * 3 + 0];
  const float cy = newxyz[(size_t)g * 3 + 1];
  const float cz = newxyz[(size_t)g * 3 + 2];

  // ---- ball query: first KK in-radius neighbors in ascending point order ----
  nbr[w][lane] = 0;
  int cnt = 0;
  const float* bxyz = xyz + (size_t)b * NN * 3;
  for (int base = 0; base < NN && cnt < KK; base += 32) {
    const int n = base + lane;
    const float dx = bxyz[n * 3 + 0] - cx;
    const float dy = bxyz[n * 3 + 1] - cy;
    const float dz = bxyz[n * 3 + 2] - cz;
    const float d2 = dx * dx + dy * dy + dz * dz;
    const bool hit = (d2 <= R2);
    const unsigned mask = (unsigned)__ballot(hit);
    const int pos = cnt + __popc(mask & ((1u << lane) - 1u));
    if (hit && pos < KK) nbr[w][pos] = n;
    cnt += __popc(mask);
  }
  if (cnt > KK) cnt = KK;
  __syncthreads();
  const int first = nbr[w][0];
  if (lane >= cnt) nbr[w][lane] = first;   // pad with first valid neighbor
  __syncthreads();

  // ---- group: X[k, 0:3]=local xyz, X[k, 3:67]=features, X[k,67]=0 ----
  const int idx = nbr[w][lane];
  float* Xr = &buf[0][w][lane][0];
  Xr[0] = bxyz[idx * 3 + 0] - cx;
  Xr[1] = bxyz[idx * 3 + 1] - cy;
  Xr[2] = bxyz[idx * 3 + 2] - cz;
  const float* fb = feat + (size_t)b * CIN * NN + idx;
#pragma unroll 8
  for (int c = 0; c < CIN; ++c) Xr[3 + c] = fb[(size_t)c * NN];
  Xr[67] = 0.0f;
  __syncthreads();

  // ---- layer 0: 67 -> 64 (buf0 -> buf1) ----
  mlp_layer(&buf[0][w][0][0], &buf[1][w][0][0], W0, b0, g0, bt0, 67, 64, lane);
  __syncthreads();
  // ---- layer 1: 64 -> 64 (buf1 -> buf0) ----
  mlp_layer(&buf[1][w][0][0], &buf[0][w][0][0], W1, b1, g1, bt1, 64, 64, lane);
  __syncthreads();

  // ---- layer 2: 64 -> 128, fused ReLU + max over the 32 samples ----
  const float* Xs = &buf[0][w][0][0];
  const int m0  = lane & 15;
  const int kk0 = (lane >> 4) << 1;
  for (int nt = 0; nt < 8; ++nt) {
    const int n = (nt << 4) + (lane & 15);
    v8f acc0 = {};
    v8f acc1 = {};
    for (int c = 0; c < 16; ++c) {
      const int c4 = c << 2;
      v2f a0, a1, bf;
      a0.x = Xs[m0 * 68 + c4 + kk0];
      a0.y = Xs[m0 * 68 + c4 + kk0 + 1];
      a1.x = Xs[(m0 + 16) * 68 + c4 + kk0];
      a1.y = Xs[(m0 + 16) * 68 + c4 + kk0 + 1];
      bf.x = W2[n * 64 + c4 + kk0];
      bf.y = W2[n * 64 + c4 + kk0 + 1];
      acc0 = __builtin_amdgcn_wmma_f32_16x16x4_f32(false, a0, false, bf,
                                                   (short)0, acc0, false, false);
      acc1 = __builtin_amdgcn_wmma_f32_16x16x4_f32(false, a1, false, bf,
                                                   (short)0, acc1, false, false);
    }
    const float s = g2[n] * INVSTD;
    const float tt = b2[n] * s + bt2[n];
    float vm = 0.0f;  // ReLU outputs are >= 0
#pragma unroll
    for (int r = 0; r < 8; ++r) {
      vm = fmaxf(vm, fmaxf(acc0[r] * s + tt, 0.0f));
      vm = fmaxf(vm, fmaxf(acc1[r] * s + tt, 0.0f));
    }
    vm = fmaxf(vm, __shfl_xor(vm, 16, 32));  // merge lane pairs (rows 0-7 / 8-15 halves)
    if (lane < 16) outFeat[((size_t)b * 128 + n) * PP + p] = vm;
  }
}

// ---------------------------------------------------------------------------
extern "C" void kernel_launch(void* const* d_in, const int* in_sizes, int n_in,
                              void* d_out, int out_size, void* d_ws, size_t ws_size,
                              hipStream_t stream) {
  (void)in_sizes; (void)n_in; (void)out_size; (void)d_ws; (void)ws_size;
  const float* xyz  = (const float*)d_in[0];
  const float* feat = (const float*)d_in[1];
  const float* W0  = (const float*)d_in[2];
  const float* b0  = (const float*)d_in[3];
  const float* g0  = (const float*)d_in[4];
  const float* bt0 = (const float*)d_in[5];
  const float* W1  = (const float*)d_in[6];
  const float* b1  = (const float*)d_in[7];
  const float* g1  = (const float*)d_in[8];
  const float* bt1 = (const float*)d_in[9];
  const float* W2  = (const float*)d_in[10];
  const float* b2  = (const float*)d_in[11];
  const float* g2  = (const float*)d_in[12];
  const float* bt2 = (const float*)d_in[13];

  float* out     = (float*)d_out;
  float* newxyz  = out;                             // [8,2048,3]
  float* outFeat = out + (size_t)BB * PP * 3;       // [8,128,2048]

  fps_kernel<<<BB, 1024, 0, stream>>>(xyz, newxyz);

  const int nCentroids = BB * PP;
  sa_fused<<<nCentroids / WAVES, WAVES * 32, 0, stream>>>(
      xyz, feat, W0, b0, g0, bt0, W1, b1, g1, bt1, W2, b2, g2, bt2,
      newxyz, outFeat);
}